// VSSBlock_56538949485141
// MI455X (gfx1250) — compile-verified
//
#include <hip/hip_runtime.h>
#include <hip/hip_bf16.h>

// ---------------- problem constants (from reference) ----------------
#define NSEQ   8          // B*4
#define L_SEQ  9216       // H*W
#define DM     64         // d_model
#define DI     128        // d_inner
#define DSTATE 16         // d_state
#define DTR    4          // dt_rank
#define KW     4          // conv kernel
#define CH     256        // scan chunk length
#define NC     36         // number of chunks (9216/256)

typedef float v2f __attribute__((ext_vector_type(2)));
typedef float v8f __attribute__((ext_vector_type(8)));
typedef int   i4v __attribute__((vector_size(16)));

__device__ __forceinline__ float silu_f(float x) { return x / (1.f + __expf(-x)); }
__device__ __forceinline__ float softplus_f(float x) { return x > 20.f ? x : log1pf(__expf(x)); }

// ---------------- CDNA5 async global->LDS copy (ASYNCcnt path) ------
#if defined(__has_builtin)
#if __has_builtin(__builtin_amdgcn_global_load_async_to_lds_b128)
#define HAVE_ASYNC_LDS 1
#endif
#endif

#ifdef HAVE_ASYNC_LDS
#define GLOBAL_AS __attribute__((address_space(1)))
#define LDS_AS    __attribute__((address_space(3)))
__device__ __forceinline__ void async_copy16(const float* g, float* l) {
    __builtin_amdgcn_global_load_async_to_lds_b128(
        (GLOBAL_AS i4v*)g, (LDS_AS i4v*)l, 0, 0);
}
__device__ __forceinline__ void async_wait0() {
#if __has_builtin(__builtin_amdgcn_s_wait_asynccnt)
    __builtin_amdgcn_s_wait_asynccnt(0);
#else
    asm volatile("s_wait_asynccnt 0x0" ::: "memory");
#endif
}
#endif

// =====================================================================
// Kernel A: fused LayerNorm + in_proj GEMM (64 -> 256), fp32 WMMA.
// Block: 256 threads (8 waves), tile = 64 rows of l for one sequence n.
// Output split: o<128 -> xs, o>=128 -> z.
// =====================================================================
__global__ __launch_bounds__(256) void k_ln_inproj(
    const float* __restrict__ in, const float* __restrict__ norm_w,
    const float* __restrict__ norm_b, const float* __restrict__ Wi /*256x64*/,
    float* __restrict__ xs, float* __restrict__ zz)
{
    __shared__ float xls[64][DM + 1];   // 64 x 65, conflict-free rows

    const int n  = blockIdx.y;
    const int l0 = blockIdx.x * 64;
    const int b  = n >> 2, g = n & 3;
    const float* xin = in + (size_t)(b * 256 + g * 64) * L_SEQ;
    const int t = threadIdx.x;

    // stage x tile (coalesced over l)
    for (int idx = t; idx < 64 * DM; idx += 256) {
        int d = idx >> 6;
        int l = idx & 63;
        xls[l][d] = xin[(size_t)d * L_SEQ + l0 + l];
    }
    __syncthreads();

    // layernorm: one row per thread (t < 64)
    if (t < 64) {
        float s = 0.f, s2 = 0.f;
        #pragma unroll 8
        for (int d = 0; d < DM; ++d) { float v = xls[t][d]; s += v; s2 += v * v; }
        float mu  = s * (1.f / DM);
        float var = s2 * (1.f / DM) - mu * mu;
        float r   = rsqrtf(var + 1e-6f);
        #pragma unroll 8
        for (int d = 0; d < DM; ++d)
            xls[t][d] = (xls[t][d] - mu) * r * norm_w[d] + norm_b[d];
    }
    __syncthreads();

    // GEMM: (64 x 64) @ Wi^T (64 x 256); wave w owns output strip [w*32, w*32+32)
    const int wv = t >> 5, lane = t & 31;
    const int half = lane >> 4, lr = lane & 15;

    for (int nt = 0; nt < 2; ++nt) {
        const int o0 = wv * 32 + nt * 16;
        v2f bfr[16];
        #pragma unroll
        for (int k = 0; k < 16; ++k) {
            const float* wp = Wi + (size_t)(o0 + lr) * DM + k * 4 + 2 * half;
            bfr[k].x = wp[0]; bfr[k].y = wp[1];
        }
        for (int m = 0; m < 4; ++m) {
            v8f acc = {};
            #pragma unroll
            for (int k = 0; k < 16; ++k) {
                v2f a;
                a.x = xls[m * 16 + lr][k * 4 + 2 * half + 0];
                a.y = xls[m * 16 + lr][k * 4 + 2 * half + 1];
                acc = __builtin_amdgcn_wmma_f32_16x16x4_f32(
                    false, a, false, bfr[k], (short)0, acc, false, false);
            }
            const int o = o0 + lr;
            float* dst = (o < DI) ? xs : zz;
            const int oc = (o < DI) ? o : (o - DI);
            #pragma unroll
            for (int r = 0; r < 8; ++r) {
                int l = l0 + m * 16 + r + 8 * half;
                dst[((size_t)n * L_SEQ + l) * DI + oc] = acc[r];
            }
        }
    }
}

// =====================================================================
// Kernel B: causal depthwise conv (K=4) + bias + SiLU. Pure bandwidth.
// float4 lanes: 16B global loads/stores.
// =====================================================================
__global__ __launch_bounds__(256) void k_conv(
    const float* __restrict__ xs, const float* __restrict__ cw /*128x4*/,
    const float* __restrict__ cb, float* __restrict__ xc)
{
    size_t idx = (size_t)blockIdx.x * 256 + threadIdx.x;   // over NLD/4
    if (idx >= (size_t)NSEQ * L_SEQ * DI / 4) return;
    const int c4 = (int)(idx & 31);          // which float4 within DI
    const size_t nl = idx >> 5;
    const int l = (int)(nl % L_SEQ);
    const int n = (int)(nl / L_SEQ);
    const int d0 = c4 * 4;

    float acc0 = cb[d0 + 0], acc1 = cb[d0 + 1], acc2 = cb[d0 + 2], acc3 = cb[d0 + 3];
    #pragma unroll
    for (int k = 0; k < KW; ++k) {
        int ls = l - (KW - 1) + k;
        if (ls >= 0) {
            const float4 v = *(const float4*)(xs + ((size_t)n * L_SEQ + ls) * DI + d0);
            acc0 += v.x * cw[(d0 + 0) * KW + k];
            acc1 += v.y * cw[(d0 + 1) * KW + k];
            acc2 += v.z * cw[(d0 + 2) * KW + k];
            acc3 += v.w * cw[(d0 + 3) * KW + k];
        }
    }
    float4 o;
    o.x = silu_f(acc0); o.y = silu_f(acc1); o.z = silu_f(acc2); o.w = silu_f(acc3);
    *(float4*)(xc + idx * 4) = o;
}

// =====================================================================
// Kernel C: x_proj (128 -> 36), split dt/B/C; dt_proj (4 -> 128) + softplus.
// Small K dims -> VALU via LDS-staged operands.
// =====================================================================
#define TC 32
__global__ __launch_bounds__(256) void k_xproj_dt(
    const float* __restrict__ xc, const float* __restrict__ xpw /*36x128*/,
    const float* __restrict__ dtw /*128x4*/, const float* __restrict__ dtbv /*128*/,
    float* __restrict__ dtbuf, float* __restrict__ Bm, float* __restrict__ Cm)
{
    __shared__ float xcl[TC][DI];
    __shared__ float xdb[TC][40];
    __shared__ float wls[36][DI];
    __shared__ float dwl[DI][4];

    const int n = blockIdx.y;
    const int l0 = blockIdx.x * TC;
    const int t = threadIdx.x;

    for (int idx = t; idx < TC * DI; idx += 256) {
        int l = idx >> 7, c = idx & 127;
        xcl[l][c] = xc[((size_t)n * L_SEQ + l0 + l) * DI + c];
    }
    for (int idx = t; idx < 36 * DI; idx += 256) wls[idx >> 7][idx & 127] = xpw[idx];
    for (int idx = t; idx < DI * 4; idx += 256)  dwl[idx >> 2][idx & 3]   = dtw[idx];
    __syncthreads();

    // x_dbl = xc @ xpw^T
    for (int idx = t; idx < TC * 36; idx += 256) {
        int l = idx / 36, o = idx % 36;
        float acc = 0.f;
        #pragma unroll 8
        for (int c = 0; c < DI; ++c) acc += xcl[l][c] * wls[o][c];
        xdb[l][o] = acc;
    }
    __syncthreads();

    // dt = softplus(dt_raw @ dtw^T + dtb)
    for (int idx = t; idx < TC * DI; idx += 256) {
        int l = idx >> 7, d = idx & 127;
        float acc = dtbv[d];
        #pragma unroll
        for (int r = 0; r < DTR; ++r) acc += xdb[l][r] * dwl[d][r];
        dtbuf[((size_t)n * L_SEQ + l0 + l) * DI + d] = softplus_f(acc);
    }
    // B, C slices
    for (int idx = t; idx < TC * DSTATE; idx += 256) {
        int l = idx >> 4, s = idx & 15;
        size_t off = ((size_t)n * L_SEQ + l0 + l) * DSTATE + s;
        Bm[off] = xdb[l][DTR + s];
        Cm[off] = xdb[l][DTR + DSTATE + s];
    }
}

// =====================================================================
// Kernel D1: scan phase 1 — per-chunk (P = prod dA, H = local end state).
// Block = (16 d) x (16 s); grid = (d_blocks=8, chunks=36, n=8).
// Tiles staged with GLOBAL_LOAD_ASYNC_TO_LDS_B128 when available.
// =====================================================================
__global__ __launch_bounds__(256) void k_scan1(
    const float* __restrict__ dt, const float* __restrict__ xc,
    const float* __restrict__ Bm, const float* __restrict__ A_log,
    float* __restrict__ Pc, float* __restrict__ Hc)
{
    __shared__ float dts[CH][16];
    __shared__ float xcs[CH][16];
    __shared__ float bms[CH][16];

    const int db = blockIdx.x, c = blockIdx.y, n = blockIdx.z;
    const int t = threadIdx.x;
    const int l0 = c * CH;

#ifdef HAVE_ASYNC_LDS
    for (int idx = t; idx < CH * 4; idx += 256) {   // 4 x b128 per 64B row
        int l = idx >> 2, q = (idx & 3) * 4;
        size_t nl = (size_t)n * L_SEQ + l0 + l;
        async_copy16(dt + nl * DI + db * 16 + q, &dts[l][q]);
        async_copy16(xc + nl * DI + db * 16 + q, &xcs[l][q]);
        async_copy16(Bm + nl * DSTATE + q, &bms[l][q]);
    }
    async_wait0();
    __syncthreads();
#else
    for (int idx = t; idx < CH * 16; idx += 256) {
        int l = idx >> 4, j = idx & 15;
        size_t nl = (size_t)n * L_SEQ + l0 + l;
        dts[l][j] = dt[nl * DI + db * 16 + j];
        xcs[l][j] = xc[nl * DI + db * 16 + j];
        bms[l][j] = Bm[nl * DSTATE + j];
    }
    __syncthreads();
#endif

    const int s = t & 15, dy = t >> 4;
    const int d = db * 16 + dy;
    const float A = -__expf(A_log[d * DSTATE + s]);
    float h = 0.f, p = 1.f;
    for (int l = 0; l < CH; ++l) {
        float dtv = dts[l][dy];
        float dA  = __expf(dtv * A);
        h = dA * h + dtv * xcs[l][dy] * bms[l][s];
        p *= dA;
    }
    size_t off = ((size_t)(n * NC + c)) * (DI * DSTATE) + d * DSTATE + s;
    Pc[off] = p;
    Hc[off] = h;
}

// =====================================================================
// Kernel D2: carry recurrence across 36 chunks (per (n,d,s)).
// =====================================================================
__global__ __launch_bounds__(256) void k_scan2(
    const float* __restrict__ Pc, const float* __restrict__ Hc,
    float* __restrict__ Hinit)
{
    int idx = blockIdx.x * 256 + threadIdx.x;   // over NSEQ * 2048 = 16384
    int n = idx >> 11;
    int j = idx & 2047;
    float h = 0.f;
    for (int c = 0; c < NC; ++c) {
        size_t off = ((size_t)(n * NC + c)) * (DI * DSTATE) + j;
        Hinit[off] = h;
        h = Pc[off] * h + Hc[off];
    }
}

// =====================================================================
// Kernel D3: scan phase 2 — replay chunk from Hinit, contract over s
// (shfl_xor width-16 within wave32 halves), + D*xc, gate with silu(z).
// =====================================================================
__global__ __launch_bounds__(256) void k_scan3(
    const float* __restrict__ dt, const float* __restrict__ xc,
    const float* __restrict__ Bm, const float* __restrict__ Cm,
    const float* __restrict__ zz, const float* __restrict__ A_log,
    const float* __restrict__ Dp, const float* __restrict__ Hinit,
    float* __restrict__ ydi)
{
    __shared__ float dts[CH][16];
    __shared__ float xcs[CH][16];
    __shared__ float bms[CH][16];

    const int db = blockIdx.x, c = blockIdx.y, n = blockIdx.z;
    const int t = threadIdx.x;
    const int l0 = c * CH;

#ifdef HAVE_ASYNC_LDS
    for (int idx = t; idx < CH * 4; idx += 256) {
        int l = idx >> 2, q = (idx & 3) * 4;
        size_t nl = (size_t)n * L_SEQ + l0 + l;
        async_copy16(dt + nl * DI + db * 16 + q, &dts[l][q]);
        async_copy16(xc + nl * DI + db * 16 + q, &xcs[l][q]);
        async_copy16(Bm + nl * DSTATE + q, &bms[l][q]);
    }
#else
    for (int idx = t; idx < CH * 16; idx += 256) {
        int l = idx >> 4, j = idx & 15;
        size_t nl = (size_t)n * L_SEQ + l0 + l;
        dts[l][j] = dt[nl * DI + db * 16 + j];
        xcs[l][j] = xc[nl * DI + db * 16 + j];
        bms[l][j] = Bm[nl * DSTATE + j];
    }
#endif
    // warm L2/L0 for the Cm stream read inside the sequential loop
    for (int l = t; l < CH; l += 256)
        __builtin_prefetch(Cm + ((size_t)n * L_SEQ + l0 + l) * DSTATE, 0, 1);
#ifdef HAVE_ASYNC_LDS
    async_wait0();
#endif
    __syncthreads();

    const int s = t & 15, dy = t >> 4;
    const int d = db * 16 + dy;
    const float A = -__expf(A_log[d * DSTATE + s]);
    size_t off = ((size_t)(n * NC + c)) * (DI * DSTATE) + d * DSTATE + s;
    float h  = Hinit[off];
    const float Dd = Dp[d];

    for (int l = 0; l < CH; ++l) {
        size_t nl = (size_t)n * L_SEQ + l0 + l;
        float dtv = dts[l][dy];
        float xv  = xcs[l][dy];
        float dA  = __expf(dtv * A);
        h = dA * h + dtv * xv * bms[l][s];
        float contrib = h * Cm[nl * DSTATE + s];
        contrib += __shfl_xor(contrib, 1, 16);
        contrib += __shfl_xor(contrib, 2, 16);
        contrib += __shfl_xor(contrib, 4, 16);
        contrib += __shfl_xor(contrib, 8, 16);
        if (s == 0) {
            float y = contrib + Dd * xv;
            ydi[nl * DI + d] = y * silu_f(zz[nl * DI + d]);
        }
    }
}

// =====================================================================
// Kernel E: out_proj GEMM (128 -> 64) + residual, fp32 WMMA.
// Writes back to (B, C, H, W) layout.
// =====================================================================
__global__ __launch_bounds__(256) void k_outproj(
    const float* __restrict__ in, const float* __restrict__ ydi,
    const float* __restrict__ Wo /*64x128*/, float* __restrict__ out)
{
    __shared__ float yls[64][DI + 1];   // 64 x 129

    const int n = blockIdx.y, l0 = blockIdx.x * 64;
    const int b = n >> 2, g = n & 3;
    const int t = threadIdx.x;

    for (int idx = t; idx < 64 * DI; idx += 256) {
        int l = idx >> 7, c = idx & 127;
        yls[l][c] = ydi[((size_t)n * L_SEQ + l0 + l) * DI + c];
    }
    __syncthreads();

    const int wv = t >> 5, lane = t & 31;
    const int half = lane >> 4, lr = lane & 15;

    // 64x64 output tile = 4 (M) x 4 (N) wmma tiles; each wave owns 2
    for (int ti = wv * 2; ti < wv * 2 + 2; ++ti) {
        const int m = ti >> 2, nt = ti & 3;
        const int o0 = nt * 16;
        v8f acc = {};
        #pragma unroll
        for (int k = 0; k < 32; ++k) {
            v2f a, bv;
            a.x = yls[m * 16 + lr][k * 4 + 2 * half + 0];
            a.y = yls[m * 16 + lr][k * 4 + 2 * half + 1];
            const float* wp = Wo + (size_t)(o0 + lr) * DI + k * 4 + 2 * half;
            bv.x = wp[0]; bv.y = wp[1];
            acc = __builtin_amdgcn_wmma_f32_16x16x4_f32(
                false, a, false, bv, (short)0, acc, false, false);
        }
        const int o = o0 + lr;
        const float* resp = in  + ((size_t)(b * 256) + g * 64 + o) * L_SEQ;
        float*       outp = out + ((size_t)(b * 256) + g * 64 + o) * L_SEQ;
        #pragma unroll
        for (int r = 0; r < 8; ++r) {
            int l = l0 + m * 16 + r + 8 * half;
            outp[l] = resp[l] + acc[r];
        }
    }
}

// =====================================================================
extern "C" void kernel_launch(void* const* d_in, const int* in_sizes, int n_in,
                              void* d_out, int out_size, void* d_ws, size_t ws_size,
                              hipStream_t stream)
{
    const float* in        = (const float*)d_in[0];
    const float* norm_w    = (const float*)d_in[1];
    const float* norm_b    = (const float*)d_in[2];
    const float* in_proj_w = (const float*)d_in[3];
    const float* conv_w    = (const float*)d_in[4];
    const float* conv_b    = (const float*)d_in[5];
    const float* x_proj_w  = (const float*)d_in[6];
    const float* dt_proj_w = (const float*)d_in[7];
    const float* dt_proj_b = (const float*)d_in[8];
    const float* A_log     = (const float*)d_in[9];
    const float* D_param   = (const float*)d_in[10];
    const float* out_proj_w= (const float*)d_in[11];

    float* ws = (float*)d_ws;
    const size_t NLD = (size_t)NSEQ * L_SEQ * DI;       // 9.44M floats
    const size_t NLS = (size_t)NSEQ * L_SEQ * DSTATE;   // 1.18M floats
    const size_t NCD = (size_t)NSEQ * NC * DI * DSTATE; // 0.59M floats

    float* xs    = ws;             // reused as ydi after conv consumes it
    float* zz    = xs + NLD;
    float* xc    = zz + NLD;
    float* dtb   = xc + NLD;
    float* Bm    = dtb + NLD;
    float* Cm    = Bm + NLS;
    float* Pc    = Cm + NLS;
    float* Hc    = Pc + NCD;
    float* Hinit = Hc + NCD;

    dim3 gA(L_SEQ / 64, NSEQ);
    k_ln_inproj<<<gA, 256, 0, stream>>>(in, norm_w, norm_b, in_proj_w, xs, zz);

    size_t tot4 = NLD / 4;
    k_conv<<<(unsigned)((tot4 + 255) / 256), 256, 0, stream>>>(xs, conv_w, conv_b, xc);

    dim3 gC(L_SEQ / TC, NSEQ);
    k_xproj_dt<<<gC, 256, 0, stream>>>(xc, x_proj_w, dt_proj_w, dt_proj_b, dtb, Bm, Cm);

    dim3 gD(DI / 16, NC, NSEQ);
    k_scan1<<<gD, 256, 0, stream>>>(dtb, xc, Bm, A_log, Pc, Hc);

    k_scan2<<<(NSEQ * DI * DSTATE) / 256, 256, 0, stream>>>(Pc, Hc, Hinit);

    float* ydi = xs;  // xs dead after k_conv
    k_scan3<<<gD, 256, 0, stream>>>(dtb, xc, Bm, Cm, zz, A_log, D_param, Hinit, ydi);

    dim3 gE(L_SEQ / 64, NSEQ);
    k_outproj<<<gE, 256, 0, stream>>>(in, ydi, out_proj_w, (float*)d_out);
}